// PositionLinear_65884798321136
// MI455X (gfx1250) — compile-verified
//
#include <hip/hip_runtime.h>
#include <hip/hip_bf16.h>

typedef float v2f __attribute__((ext_vector_type(2)));
typedef float v4f __attribute__((ext_vector_type(4)));
typedef float v8f __attribute__((ext_vector_type(8)));

#define N_FEAT   64
#define TOKENS   (32 * 32768)        // B * P
#define TILES    (TOKENS / 16)       // 65536 tiles of 16 tokens

// out[m, n] = relu( sum_k x[m,k] * W[n,k] + b[n] )
// Wave computes a 16(M) x 64(N) tile via V_WMMA_F32_16X16X4_F32.
// Logical K mapping per WMMA step kk: slots {0,1} = k{2kk,2kk+1} (lanes 0-15),
// slots {2,3} = k{32+2kk,33+2kk} (lanes 16-31) -> contiguous 128B per lane.
__global__ __launch_bounds__(256) void poslinear_wmma_f32(
    const float* __restrict__ x,
    const float* __restrict__ W,
    const float* __restrict__ b,
    float* __restrict__ out)
{
    const int lane  = threadIdx.x & 31;
    const int half  = lane >> 4;      // 0: k in [0,32), 1: k in [32,64)
    const int mrow  = lane & 15;      // row within tile (A), N within tile (B/C/D)
    const int waveId = (int)((blockIdx.x * blockDim.x + threadIdx.x) >> 5);
    const int nWaves = (int)((gridDim.x * blockDim.x) >> 5);

    // ---- Preload W fragments (register-resident for the whole kernel) ----
    // wreg[nt][j] = W[(nt*16 + mrow)*64 + half*32 + j], j = 0..31
    float wreg[4][32];
#pragma unroll
    for (int nt = 0; nt < 4; ++nt) {
        const float* wrow = W + (nt * 16 + mrow) * N_FEAT + half * 32;
#pragma unroll
        for (int j = 0; j < 8; ++j) {
            v4f v = *(const v4f*)(wrow + 4 * j);
            wreg[nt][4 * j + 0] = v[0];
            wreg[nt][4 * j + 1] = v[1];
            wreg[nt][4 * j + 2] = v[2];
            wreg[nt][4 * j + 3] = v[3];
        }
    }
    // Bias per N-tile: C/D layout puts N = mrow for this lane in every acc VGPR.
    float biasv[4];
#pragma unroll
    for (int nt = 0; nt < 4; ++nt) biasv[nt] = b[nt * 16 + mrow];

    // ---- Persistent loop over 16-token tiles (uniform per wave) ----
    for (int tile = waveId; tile < TILES; tile += nWaves) {
        // A fragments: areg[j] = x[(tile*16 + mrow)*64 + half*32 + j]
        float areg[32];
        const float* xrow = x + (size_t)(tile * 16 + mrow) * N_FEAT + half * 32;
#pragma unroll
        for (int j = 0; j < 8; ++j) {
            v4f v = *(const v4f*)(xrow + 4 * j);
            areg[4 * j + 0] = v[0];
            areg[4 * j + 1] = v[1];
            areg[4 * j + 2] = v[2];
            areg[4 * j + 3] = v[3];
        }

        // Init accumulators with bias (C layout: same bias for all 8 rows/lane).
        v8f acc[4];
#pragma unroll
        for (int nt = 0; nt < 4; ++nt)
#pragma unroll
            for (int e = 0; e < 8; ++e) acc[nt][e] = biasv[nt];

        // 16 K-steps x 4 N-tiles = 64 WMMAs; 4 independent acc chains interleaved
        // so each chain's D->C dependency is separated by 3 other WMMAs.
#pragma unroll
        for (int kk = 0; kk < 16; ++kk) {
            v2f a;
            a[0] = areg[2 * kk + 0];
            a[1] = areg[2 * kk + 1];
#pragma unroll
            for (int nt = 0; nt < 4; ++nt) {
                v2f bb;
                bb[0] = wreg[nt][2 * kk + 0];
                bb[1] = wreg[nt][2 * kk + 1];
                acc[nt] = __builtin_amdgcn_wmma_f32_16x16x4_f32(
                    /*neg_a=*/false, a,
                    /*neg_b=*/false, bb,
                    /*c_mod=*/(short)0, acc[nt],
                    /*reuse_a=*/false, /*reuse_b=*/false);
            }
        }

        // Epilogue: ReLU + store. acc[nt] VGPR i, this lane ->
        //   token = tile*16 + i + 8*half, col n = nt*16 + mrow.
#pragma unroll
        for (int nt = 0; nt < 4; ++nt) {
#pragma unroll
            for (int i = 0; i < 8; ++i) {
                float v = acc[nt][i];
                v = fmaxf(v, 0.0f);
                out[(size_t)(tile * 16 + i + 8 * half) * N_FEAT + nt * 16 + mrow] = v;
            }
        }
    }
}

extern "C" void kernel_launch(void* const* d_in, const int* in_sizes, int n_in,
                              void* d_out, int out_size, void* d_ws, size_t ws_size,
                              hipStream_t stream) {
    (void)in_sizes; (void)n_in; (void)out_size; (void)d_ws; (void)ws_size;
    const float* x = (const float*)d_in[0];   // (32, 32768, 64) f32
    const float* W = (const float*)d_in[1];   // (64, 64) f32
    const float* b = (const float*)d_in[2];   // (64,) f32
    float* out = (float*)d_out;               // (32, 32768, 64) f32

    // 1024 blocks x 256 threads = 8192 waves; 65536 tiles -> exactly 8 tiles/wave,
    // fully uniform grid-stride loop (EXEC stays all-ones for WMMA).
    dim3 grid(1024), block(256);
    poslinear_wmma_f32<<<grid, block, 0, stream>>>(x, W, b, out);
}